// LightCurveFolder_18794776887586
// MI455X (gfx1250) — compile-verified
//
#include <hip/hip_runtime.h>
#include <hip/hip_bf16.h>
#include <math.h>

#define S       4096
#define NT      256
#define NI      50
#define FLOATMAX 3.402823466e+38f
// exp((cos-1)/BW^2) = exp2((cos-1) * 100 * log2(e))
#define KEXP    144.26950408889634f

typedef float v2f __attribute__((ext_vector_type(2)));
typedef float v8f __attribute__((ext_vector_type(8)));

__device__ __forceinline__ float circ_delta(float d) {
    // remainder(d - 0.5, 1.0) - 0.5
    float f = d - 0.5f;
    return f - floorf(f) - 0.5f;
}

// In-LDS bitonic sort of (key, idx) pairs, ascending. 4096 elems, 256 threads.
__device__ __forceinline__ void bitonic_sort(float* key, unsigned short* idx, int tid) {
    for (int k = 2; k <= S; k <<= 1) {
        for (int j = k >> 1; j > 0; j >>= 1) {
            for (int t = tid; t < S; t += NT) {
                int ixj = t ^ j;
                if (ixj > t) {
                    float a = key[t];
                    float c = key[ixj];
                    bool up = ((t & k) == 0);
                    bool sw = up ? (a > c) : (a < c);
                    if (sw) {
                        key[t] = c; key[ixj] = a;
                        unsigned short ia = idx[t];
                        idx[t] = idx[ixj];
                        idx[ixj] = ia;
                    }
                }
            }
            __syncthreads();
        }
    }
}

__global__ __launch_bounds__(NT) void lightcurve_fold_kernel(
    const float* __restrict__ lc,
    const unsigned char* __restrict__ validg,
    const float* __restrict__ freqg,
    float* __restrict__ out_lc,
    float* __restrict__ out_valid) {

    __shared__ float          s_phase[S];      // 16 KB  original-order phase (later shifted)
    __shared__ float          s_mag[S];        // 16 KB  original-order mag
    __shared__ float          s_key[S];        // 16 KB  sort keys
    __shared__ unsigned short s_idx[S];        //  8 KB  sorted-pos -> original index
    __shared__ unsigned char  s_valid[S];      //  4 KB
    __shared__ float          s_kde[2][64][2]; //  1 KB  [khalf][interp row][num/den]
    __shared__ float          s_smooth[NI];
    __shared__ float          s_shift;

    const int b   = blockIdx.x;
    const int tid = threadIdx.x;

    const float* timep = lc + (size_t)b * 3 * S;
    const float* magp  = timep + S;
    const float* errp  = magp + S;
    const float  invP  = freqg[b] * 0.5f;   // 1/period, period = 2/freq

    // ---- Stage 0a: async global->LDS copies (mag row + valid row), ASYNCcnt-tracked ----
    {
        const unsigned mag_lds   = (unsigned)(uintptr_t)&s_mag[0];
        const unsigned valid_lds = (unsigned)(uintptr_t)&s_valid[0];
        const unsigned char* vrow = validg + (size_t)b * S;
        for (int t = tid; t < S; t += NT) {
            unsigned ldsa = mag_lds + (unsigned)t * 4u;
            unsigned voff = (unsigned)t * 4u;
            asm volatile("global_load_async_to_lds_b32 %0, %1, %2"
                         :: "v"(ldsa), "v"(voff), "s"(magp) : "memory");
        }
        for (int t = tid; t < S / 4; t += NT) {   // 4096 bool bytes = 1024 words
            unsigned ldsa = valid_lds + (unsigned)t * 4u;
            unsigned voff = (unsigned)t * 4u;
            asm volatile("global_load_async_to_lds_b32 %0, %1, %2"
                         :: "v"(ldsa), "v"(voff), "s"(vrow) : "memory");
        }
    }

    // ---- Stage 0b: load time, fold phase (overlaps the async copies) ----
    for (int t = tid; t < S; t += NT) {
        float tt = timep[t];
        float r  = tt * invP;
        float ph = r - floorf(r);            // remainder(time, P)/P
        s_phase[t] = ph;
        s_key[t]   = ph;
        s_idx[t]   = (unsigned short)t;
        if ((t & 31) == 0) __builtin_prefetch(errp + t, 0, 0);  // warm L2 for final gather
    }
    asm volatile("s_wait_asynccnt 0x0" ::: "memory");
    __syncthreads();

    // ---- Stage 1: sort by phase ----
    bitonic_sort(s_key, s_idx, tid);

    // ---- Stage 2: circular windowed outlier threshold (w=5, center excluded) ----
    for (int t = tid; t < S; t += NT) {
        float pc = s_key[t];
        int jm2 = (t - 2) & (S - 1), jm1 = (t - 1) & (S - 1);
        int jp1 = (t + 1) & (S - 1), jp2 = (t + 2) & (S - 1);
        float d0 = circ_delta(s_key[jm2] - pc);
        float d1 = circ_delta(s_key[jm1] - pc);
        float d2 = circ_delta(s_key[jp1] - pc);
        float d3 = circ_delta(s_key[jp2] - pc);
        float w0 = 0.75f * (1.0f - d0 * d0);
        float w1 = 0.75f * (1.0f - d1 * d1);
        float w2 = 0.75f * (1.0f - d2 * d2);
        float w3 = 0.75f * (1.0f - d3 * d3);
        float q0 = s_mag[s_idx[jm2]];
        float q1 = s_mag[s_idx[jm1]];
        float q2 = s_mag[s_idx[jp1]];
        float q3 = s_mag[s_idx[jp2]];
        float ws  = w0 + w1 + w2 + w3;
        float loc = (q0 * w0 + q1 * w1 + q2 * w2 + q3 * w3) / ws;
        float e0 = q0 - loc, e1 = q1 - loc, e2 = q2 - loc, e3 = q3 - loc;
        float var = (e0 * e0 * w0 + e1 * e1 * w1 + e2 * e2 * w2 + e3 * e3 * w3) / (ws - 1.0f);
        float thr = loc + 5.0f * sqrtf(var);
        int oi = s_idx[t];
        if (s_mag[oi] > thr) s_valid[oi] = 0;   // faint outlier -> invalid
    }
    __syncthreads();

    // ---- Stage 3: KDE smoothing as fp32 WMMA GEMM ----
    // M = 64 (50 interp rows padded), N = 16 (col0 = Σ K*mag, col1 = Σ K), K = 4096.
    // 8 waves: 4 M-tiles x 2 K-halves; A computed on the fly, accumulate in f32 C.
    // Fully branchless inner loop: masks are float multipliers, B built with FMA selectors.
    {
        const int lane  = tid & 31;
        const int wave  = tid >> 5;
        const int tile  = wave & 3;          // interp rows tile*16 .. +15
        const int khalf = wave >> 2;         // K range [khalf*2048, +2048)
        const int mi    = tile * 16 + (lane & 15);
        const float mokf = (mi < NI) ? 1.0f : 0.0f;          // row mask (padded rows -> 0)
        const float phim = (mi < NI) ? (float)mi * (1.0f / 49.0f) : 0.0f;
        const int  kO   = (lane >> 4) * 2;   // A/B: K=0,1 in lanes 0-15; K=2,3 in lanes 16-31
        const int  n    = lane & 15;
        const float selM   = (n == 0) ? 1.0f : 0.0f;         // B col 0 = mag
        const float selOne = (n == 1) ? 1.0f : 0.0f;         // B col 1 = 1

        v8f c = {};
        const int kbase = khalf * 2048 + kO;
        #pragma unroll 4
        for (int kb = 0; kb < 2048; kb += 4) {
            int k0 = kbase + kb;                              // even -> 8B aligned
            float2 ph = *(const float2*)&s_phase[k0];
            float2 mg = *(const float2*)&s_mag[k0];
            unsigned vp = *(const unsigned short*)&s_valid[k0];
            // hardware cos takes revolutions: v_cos(x) = cos(2*pi*x)
            float kv0 = __builtin_amdgcn_exp2f((__builtin_amdgcn_cosf(phim - ph.x) - 1.0f) * KEXP);
            float kv1 = __builtin_amdgcn_exp2f((__builtin_amdgcn_cosf(phim - ph.y) - 1.0f) * KEXP);
            float a0 = kv0 * ((float)(vp & 0xffu) * mokf);
            float a1 = kv1 * ((float)(vp >> 8)    * mokf);
            float b0 = fmaf(mg.x, selM, selOne);
            float b1 = fmaf(mg.y, selM, selOne);
            v2f A;  A.x = a0;  A.y = a1;
            v2f Bm; Bm.x = b0; Bm.y = b1;
            c = __builtin_amdgcn_wmma_f32_16x16x4_f32(
                    /*neg_a=*/false, A, /*neg_b=*/false, Bm,
                    /*c_mod=*/(short)0, c, /*reuse_a=*/false, /*reuse_b=*/false);
        }
        // C/D layout: lanes 0-15 hold M = r, lanes 16-31 hold M = r + 8, N = lane & 15
        if (n < 2) {
            int mhi = (lane >> 4) * 8;
            #pragma unroll
            for (int r = 0; r < 8; ++r) {
                s_kde[khalf][tile * 16 + mhi + r][n] = c[r];
            }
        }
    }
    __syncthreads();

    if (tid < NI) {
        float num = s_kde[0][tid][0] + s_kde[1][tid][0];
        float den = s_kde[0][tid][1] + s_kde[1][tid][1];
        s_smooth[tid] = num / den;
    }
    __syncthreads();
    if (tid == 0) {
        float best = s_smooth[0];
        int bi = 0;
        for (int i = 1; i < NI; ++i)
            if (s_smooth[i] > best) { best = s_smooth[i]; bi = i; }  // first max, like argmax
        s_shift = (float)bi * (1.0f / 49.0f);
    }
    __syncthreads();

    // ---- Stage 4: shift phase, sort valid-first by phase ----
    const float shift = s_shift;
    for (int t = tid; t < S; t += NT) {
        float ph = s_phase[t] - shift;
        s_phase[t] = ph;
        s_key[t]   = s_valid[t] ? ph : FLOATMAX;
        s_idx[t]   = (unsigned short)t;
    }
    __syncthreads();

    bitonic_sort(s_key, s_idx, tid);

    // ---- Stage 5: gather outputs ----
    for (int t = tid; t < S; t += NT) {
        int oi = s_idx[t];
        out_lc[((size_t)b * 3 + 0) * S + t] = s_phase[oi];
        out_lc[((size_t)b * 3 + 1) * S + t] = s_mag[oi];
        out_lc[((size_t)b * 3 + 2) * S + t] = errp[oi];
        out_valid[(size_t)b * S + t] = s_valid[oi] ? 1.0f : 0.0f;
    }
}

extern "C" void kernel_launch(void* const* d_in, const int* in_sizes, int n_in,
                              void* d_out, int out_size, void* d_ws, size_t ws_size,
                              hipStream_t stream) {
    const float*         lc    = (const float*)d_in[0];          // [B,3,S] f32
    const unsigned char* valid = (const unsigned char*)d_in[1];  // [B,S] bool (1 byte)
    const float*         freq  = (const float*)d_in[2];          // [B] f32

    const int B = in_sizes[0] / (3 * S);
    float* out_lc    = (float*)d_out;                 // [B,3,S]
    float* out_valid = out_lc + (size_t)B * 3 * S;    // [B,S]

    if (B > 0) {
        lightcurve_fold_kernel<<<B, NT, 0, stream>>>(lc, valid, freq, out_lc, out_valid);
    }
}